// HOIModel_27023934227209
// MI455X (gfx1250) — compile-verified
//
#include <hip/hip_runtime.h>
#include <hip/hip_bf16.h>
#include <math.h>

typedef __attribute__((ext_vector_type(2))) float v2f;
typedef __attribute__((ext_vector_type(8))) float v8f;

#define NL 8

__device__ __forceinline__ float wave_reduce_add(float v) {
    #pragma unroll
    for (int off = 16; off > 0; off >>= 1) v += __shfl_xor(v, off, 32);
    return v;
}

__device__ __forceinline__ void load_row(const float* __restrict__ P, int idx, float* r) {
    const float4* p = (const float4*)(P + (size_t)idx * NL);
    float4 a = p[0], b = p[1];
    r[0]=a.x; r[1]=a.y; r[2]=a.z; r[3]=a.w; r[4]=b.x; r[5]=b.y; r[6]=b.z; r[7]=b.w;
}

__global__ void zero_out_kernel(float* out) {
    if (threadIdx.x == 0 && blockIdx.x == 0) out[0] = 0.0f;
}

// Scatter + row softmax: P[idx[i]] = softmax(params[i])
__global__ void softmax_scatter_kernel(const float* __restrict__ fixedP,
                                       const float* __restrict__ trainP,
                                       const int*   __restrict__ fixedIdx,
                                       const int*   __restrict__ trainIdx,
                                       float* __restrict__ P,
                                       int nFixed, int nTotal) {
    int i = blockIdx.x * blockDim.x + threadIdx.x;
    if (i >= nTotal) return;
    const float* src; int dst;
    if (i < nFixed) { src = fixedP + (size_t)i * NL;            dst = fixedIdx[i]; }
    else            { src = trainP + (size_t)(i - nFixed) * NL; dst = trainIdx[i - nFixed]; }
    const float4* s4 = (const float4*)src;
    float4 lo = s4[0], hi = s4[1];
    float x[8] = {lo.x, lo.y, lo.z, lo.w, hi.x, hi.y, hi.z, hi.w};
    float mx = x[0];
    #pragma unroll
    for (int j = 1; j < 8; ++j) mx = fmaxf(mx, x[j]);
    float s = 0.f;
    #pragma unroll
    for (int j = 0; j < 8; ++j) { x[j] = __expf(x[j] - mx); s += x[j]; }
    float inv = 1.0f / s;
    float4* d = (float4*)(P + (size_t)dst * NL);
    d[0] = make_float4(x[0]*inv, x[1]*inv, x[2]*inv, x[3]*inv);
    d[1] = make_float4(x[4]*inv, x[5]*inv, x[6]*inv, x[7]*inv);
}

// pairs term = <coef2(8x8), U^T V>, K-accumulated with V_WMMA_F32_16X16X4_F32.
// Two independent pair sets are packed block-diagonally into the 16x16 tile:
//   rows/cols 0-7  <- set 1 (4 pairs across K=0..3)
//   rows/cols 8-15 <- set 2 (4 pairs across K=0..3)
// Cross blocks accumulate garbage that is never read. 8 pairs per WMMA,
// all 32 lanes load real data, no guards -> straight-line body (EXEC all-1s).
// A layout (ISA 7.12.2, 32-bit 16x4): VGPR0={K0|K2}, VGPR1={K1|K3} across
// lane halves, M = lane&15. B assumed symmetric.
__global__ void pairs_wmma_kernel(const float* __restrict__ P,
                                  const int*   __restrict__ pairs,
                                  const float* __restrict__ coef2,
                                  float* __restrict__ out, int nPairs) {
    const int lane = threadIdx.x & 31;
    const int half = lane >> 4;        // 0: K slots 0/1, 1: K slots 2/3
    const int m    = lane & 15;        // M for A, N for B
    const int sel  = m >> 3;           // 0: pair set 1, 1: pair set 2
    const int comp = m & 7;            // component within 8-dim row
    const int wavesPerBlock = blockDim.x >> 5;
    const long waveId = (long)blockIdx.x * wavesPerBlock + (threadIdx.x >> 5);
    const long nWaves = (long)gridDim.x * wavesPerBlock;
    const long nBig = (long)(nPairs >> 5);          // 32-pair blocks

    const int2* __restrict__ prs = (const int2*)pairs;  // (u_idx, v_idx)

    v8f acc = {};
    for (long b = waveId; b < nBig; b += nWaves) {
        // speculative prefetch of next index block (address clamped in-bounds,
        // executed by all lanes -> no EXEC manipulation near WMMA)
        long bn = b + nWaves; if (bn >= nBig) bn = b;
        __builtin_prefetch(&prs[bn << 5], 0, 0);

        const int base = (int)(b << 5);
        v2f A[4], B[4];
        #pragma unroll
        for (int s = 0; s < 4; ++s) {
            int p0 = base + s * 8 + sel * 4 + 2 * half;  // pair for K = 2*half
            int2 pr0 = prs[p0];                          // pair for K = 2*half
            int2 pr1 = prs[p0 + 1];                      // pair for K = 2*half+1
            A[s].x = P[(size_t)pr0.x * NL + comp];
            B[s].x = P[(size_t)pr0.y * NL + comp];
            A[s].y = P[(size_t)pr1.x * NL + comp];
            B[s].y = P[(size_t)pr1.y * NL + comp];
        }
        #pragma unroll
        for (int s = 0; s < 4; ++s)
            acc = __builtin_amdgcn_wmma_f32_16x16x4_f32(
                false, A[s], false, B[s], (short)0, acc, false, false);
    }

    // Readout. D layout: lanes 0-15 -> M=r, N=lane; lanes 16-31 -> M=8+r, N=lane-16.
    // Set-1 block (M<8,N<8): lanes 0-7. Set-2 block (M,N in 8..15): lanes 24-31.
    // Both: block row = r, block col = m&7.
    float partial = 0.f;
    if (half == sel) {
        #pragma unroll
        for (int r = 0; r < 8; ++r) partial += acc[r] * coef2[r * 8 + comp];
    }
    partial = wave_reduce_add(partial);
    if (lane == 0) atomicAdd(out, partial);

    // Scalar tail for nPairs % 32 (zero for the reference shapes).
    const int tailStart = (int)(nBig << 5);
    const int nTail = nPairs - tailStart;
    if (blockIdx.x == 0 && (int)threadIdx.x < nTail) {
        int2 pr = prs[tailStart + threadIdx.x];
        float u[8], v[8];
        load_row(P, pr.x, u); load_row(P, pr.y, v);
        float s = 0.f;
        #pragma unroll
        for (int i = 0; i < 8; ++i) {
            float si = 0.f;
            #pragma unroll
            for (int j = 0; j < 8; ++j) si += coef2[i * 8 + j] * v[j];
            s += u[i] * si;
        }
        atomicAdd(out, s);
    }
}

// triples: t = sum_i u_i * sum_j v_j * dot(coef3[i][j][:], w). coef3 in LDS,
// loop indices are lane-uniform -> broadcast ds_load_b128 reads.
__global__ void triples_kernel(const float* __restrict__ P,
                               const int*   __restrict__ triples,
                               const float* __restrict__ coef3,
                               float* __restrict__ out, int nTriples) {
    __shared__ __align__(16) float c3[512];
    for (int i = threadIdx.x; i < 512; i += blockDim.x) c3[i] = coef3[i];
    __syncthreads();

    float sum = 0.f;
    const int stride = gridDim.x * blockDim.x;
    for (int n = blockIdx.x * blockDim.x + threadIdx.x; n < nTriples; n += stride) {
        int ia = triples[3*n+0], ib = triples[3*n+1], ic = triples[3*n+2];
        float u[8], v[8], w[8];
        load_row(P, ia, u); load_row(P, ib, v); load_row(P, ic, w);
        float t = 0.f;
        #pragma unroll
        for (int i = 0; i < 8; ++i) {
            float ti = 0.f;
            #pragma unroll
            for (int j = 0; j < 8; ++j) {
                const float4* cr = (const float4*)&c3[(i * 8 + j) * 8];
                float4 cA = cr[0], cB = cr[1];
                float tj = cA.x*w[0] + cA.y*w[1] + cA.z*w[2] + cA.w*w[3]
                         + cB.x*w[4] + cB.y*w[5] + cB.z*w[6] + cB.w*w[7];
                ti += v[j] * tj;
            }
            t += u[i] * ti;
        }
        sum += t;
    }
    sum = wave_reduce_add(sum);
    if ((threadIdx.x & 31) == 0) atomicAdd(out, sum);
}

// quads: Horner over 4 levels; coef4 (16KB) in LDS (320KB/WGP -> trivial fit).
__global__ void quads_kernel(const float* __restrict__ P,
                             const int*   __restrict__ quads,
                             const float* __restrict__ coef4,
                             float* __restrict__ out, int nQuads) {
    __shared__ __align__(16) float c4[4096];
    for (int i = threadIdx.x; i < 4096; i += blockDim.x) c4[i] = coef4[i];
    __syncthreads();

    float sum = 0.f;
    const int stride = gridDim.x * blockDim.x;
    for (int n = blockIdx.x * blockDim.x + threadIdx.x; n < nQuads; n += stride) {
        int ia = quads[4*n+0], ib = quads[4*n+1], ic = quads[4*n+2], id = quads[4*n+3];
        float u[8], v[8], w[8], x[8];
        load_row(P, ia, u); load_row(P, ib, v); load_row(P, ic, w); load_row(P, id, x);
        float t = 0.f;
        #pragma unroll
        for (int i = 0; i < 8; ++i) {
            float ti = 0.f;
            #pragma unroll
            for (int j = 0; j < 8; ++j) {
                float tj = 0.f;
                #pragma unroll
                for (int k = 0; k < 8; ++k) {
                    const float4* cr = (const float4*)&c4[((i * 8 + j) * 8 + k) * 8];
                    float4 cA = cr[0], cB = cr[1];
                    float tk = cA.x*x[0] + cA.y*x[1] + cA.z*x[2] + cA.w*x[3]
                             + cB.x*x[4] + cB.y*x[5] + cB.z*x[6] + cB.w*x[7];
                    tj += w[k] * tk;
                }
                ti += v[j] * tj;
            }
            t += u[i] * ti;
        }
        sum += t;
    }
    sum = wave_reduce_add(sum);
    if ((threadIdx.x & 31) == 0) atomicAdd(out, sum);
}

extern "C" void kernel_launch(void* const* d_in, const int* in_sizes, int n_in,
                              void* d_out, int out_size, void* d_ws, size_t ws_size,
                              hipStream_t stream) {
    const float* fixedP   = (const float*)d_in[0];
    const float* trainP   = (const float*)d_in[1];
    const float* coef2    = (const float*)d_in[2];
    const float* coef3    = (const float*)d_in[3];
    const float* coef4    = (const float*)d_in[4];
    const int*   fixedIdx = (const int*)d_in[5];
    const int*   trainIdx = (const int*)d_in[6];
    const int*   pairs    = (const int*)d_in[7];
    const int*   triples  = (const int*)d_in[8];
    const int*   quads    = (const int*)d_in[9];

    const int nFixed   = in_sizes[5];
    const int nTrain   = in_sizes[6];
    const int nTotal   = nFixed + nTrain;
    const int nPairs   = in_sizes[7] / 2;
    const int nTriples = in_sizes[8] / 3;
    const int nQuads   = in_sizes[9] / 4;

    float* P   = (float*)d_ws;   // 100000 x 8 f32 = 3.2 MB scratch
    float* out = (float*)d_out;  // single f32 scalar

    zero_out_kernel<<<1, 1, 0, stream>>>(out);
    softmax_scatter_kernel<<<(nTotal + 255) / 256, 256, 0, stream>>>(
        fixedP, trainP, fixedIdx, trainIdx, P, nFixed, nTotal);
    pairs_wmma_kernel<<<512, 256, 0, stream>>>(P, pairs, coef2, out, nPairs);
    triples_kernel<<<128, 256, 0, stream>>>(P, triples, coef3, out, nTriples);
    quads_kernel<<<64, 256, 0, stream>>>(P, quads, coef4, out, nQuads);
}